// Pretrain_5MODE_29901562315447
// MI455X (gfx1250) — compile-verified
//
#include <hip/hip_runtime.h>
#include <cstdint>
#include <cstddef>

#define BINS_NUM   35
#define BINS_SIZE  9
#define NM         2001   // 9 + 45 + 165 + 495 + 1287 monomials per bin
#define TPAD       720    // LDS table slots per wave (715 used, padded)
#define PAIRS_PER_BLOCK 8
#define BLOCK      256

// Phase boundaries in output index t (lexicographic combos, degree 1..5):
//   deg1 [0,9)  deg2 [9,54)  deg3 [54,219)  deg4 [219,714)  deg5 [714,2001)
// LDS table T layout: T[0]=1.0, T[1..9]=v, T[10..54]=deg2, T[55..219]=deg3,
// T[220..714]=deg4.  For any output t of degree<=4, its table slot is t+1.
// Every output t = T[parent(t)] * T[1 + last(t)] — left-associated, matching
// the reference's sequential prod() exactly.

namespace {

struct Tab { unsigned short e[NM]; };   // bits[9:0]=parent table idx, bits[15:12]=last var

constexpr Tab make_tab() {
  Tab t{};
  int n = 0;
  // deg1: parent = unity (T[0]), last = a
  for (int a = 0; a < 9; ++a)
    t.e[n++] = (unsigned short)(0 | (a << 12));
  // deg2: parent = T[1+a] (= v_a)
  for (int a = 0; a < 9; ++a)
    for (int b = a; b < 9; ++b)
      t.e[n++] = (unsigned short)((1 + a) | (b << 12));
  // deg3: parent walks the deg2 slots 10..54 in the same lex order
  { int p = 10;
    for (int a = 0; a < 9; ++a)
      for (int b = a; b < 9; ++b) {
        for (int c = b; c < 9; ++c)
          t.e[n++] = (unsigned short)(p | (c << 12));
        ++p;
      } }
  // deg4: parent walks deg3 slots 55..219
  { int p = 55;
    for (int a = 0; a < 9; ++a)
      for (int b = a; b < 9; ++b)
        for (int c = b; c < 9; ++c) {
          for (int d = c; d < 9; ++d)
            t.e[n++] = (unsigned short)(p | (d << 12));
          ++p;
        } }
  // deg5: parent walks deg4 slots 220..714
  { int p = 220;
    for (int a = 0; a < 9; ++a)
      for (int b = a; b < 9; ++b)
        for (int c = b; c < 9; ++c)
          for (int d = c; d < 9; ++d) {
            for (int f = d; f < 9; ++f)
              t.e[n++] = (unsigned short)(p | (f << 12));
            ++p;
          } }
  return t;
}

__constant__ Tab g_tab = make_tab();

} // namespace

template <bool STORE_TO_TABLE>
__device__ __forceinline__ void run_phase(int t0, int t1, int lane,
                                          const unsigned short* __restrict__ tab,
                                          float* Tw,
                                          float* __restrict__ dst) {
  for (int t = t0 + lane; t < t1; t += 32) {
    const unsigned e = tab[t];                       // coalesced u16, L2-hot 4 KB
    const float p = Tw[e & 0x3FFu] * Tw[1u + (e >> 12)];   // 2 LDS loads, 1 mul
    if (STORE_TO_TABLE) Tw[t + 1] = p;               // append for next degree
    __builtin_nontemporal_store(p, dst + t);         // 287 MB stream: bypass L2
  }
}

// One wave32 owns one (batch, bin) pair end-to-end; 8 pairs per block.
__global__ __launch_bounds__(BLOCK) void poly_expand_kernel(
    const float* __restrict__ x, float* __restrict__ out, int total_pairs) {
  const int lane = threadIdx.x & 31;
  const int wave = threadIdx.x >> 5;
  int pair = blockIdx.x * PAIRS_PER_BLOCK + wave;
  if (pair >= total_pairs) pair = total_pairs - 1;   // clamp: duplicate work, no divergent barrier

  __shared__ float T[PAIRS_PER_BLOCK][TPAD];         // 23 KB of the 320 KB WGP LDS
  float* Tw = T[wave];

  const float* src = x + (size_t)pair * BINS_SIZE;

  if (lane == 0) Tw[0] = 1.0f;                       // unity slot
  // CDNA5 async global->LDS copy: lanes 0..8 stage v into T[1..9] (ASYNCcnt).
  if (lane < BINS_SIZE) {
    unsigned lds_addr = (unsigned)(uintptr_t)&Tw[1 + lane];  // low 32b of flat = LDS offset
    unsigned long long ga = (unsigned long long)(uintptr_t)(src + lane);
    asm volatile("global_load_async_to_lds_b32 %0, %1, off"
                 :: "v"(lds_addr), "v"(ga)
                 : "memory");
  }
  asm volatile("s_wait_asynccnt 0" ::: "memory");    // issuing wave waits its own count
  __syncthreads();

  float* dst = out + (size_t)pair * NM;
  const unsigned short* tab = g_tab.e;

  run_phase<false>(  0,    9, lane, tab, Tw, dst);   // deg1: 1.0 * v_a (exact)
  __syncthreads();
  run_phase<true >(  9,   54, lane, tab, Tw, dst);   // deg2
  __syncthreads();
  run_phase<true >( 54,  219, lane, tab, Tw, dst);   // deg3
  __syncthreads();
  run_phase<true >(219,  714, lane, tab, Tw, dst);   // deg4
  __syncthreads();
  run_phase<false>(714,   NM, lane, tab, Tw, dst);   // deg5 (64% of output)
}

extern "C" void kernel_launch(void* const* d_in, const int* in_sizes, int n_in,
                              void* d_out, int out_size, void* d_ws, size_t ws_size,
                              hipStream_t stream) {
  const float* x = (const float*)d_in[0];
  float* out = (float*)d_out;
  const int B = in_sizes[0] / (BINS_NUM * BINS_SIZE);     // 1024 for the reference
  const int total_pairs = B * BINS_NUM;                   // 35840
  const int blocks = (total_pairs + PAIRS_PER_BLOCK - 1) / PAIRS_PER_BLOCK;
  poly_expand_kernel<<<dim3((unsigned)blocks), dim3(BLOCK), 0, stream>>>(x, out, total_pairs);
}